// ProductQuantizer_6021544149378
// MI455X (gfx1250) — compile-verified
//
#include <hip/hip_runtime.h>
#include <hip/hip_bf16.h>

typedef float v2f __attribute__((ext_vector_type(2)));
typedef float v8f __attribute__((ext_vector_type(8)));

#define ZDIM   512
#define GE_DIM 640
#define NGRP   2
#define NENT   320
#define VDIM   128
#define BDIM   16
#define TDIM   4096
#define MTOT   (BDIM * TDIM)   /* 65536 */

// ---------------------------------------------------------------------------
// Kernel 1: logits[m, n] = sum_k x[b, k, t] * W[n, k] + bias[n]
//   m = b*T + t (t contiguous in x's last dim -> A is M-contiguous)
//   One wave computes a 16(M) x 64(N) tile via 4 accumulators,
//   K stepped by 4 with V_WMMA_F32_16X16X4_F32.
// A layout (16x4 f32): lanes 0-15 hold K={0,1}, lanes 16-31 hold K={2,3}.
// B layout (4x16 f32): lane = N; lanes 0-15 K={0,1}, lanes 16-31 K={2,3}
//   -> B fragment is a contiguous float2 along K of W[n, :].
// ---------------------------------------------------------------------------
__global__ __launch_bounds__(256) void pq_gemm_logits(
    const float* __restrict__ x, const float* __restrict__ w,
    const float* __restrict__ bias, float* __restrict__ logits)
{
    const int lane = threadIdx.x & 31;
    const int wave = threadIdx.x >> 5;
    const int mr   = lane & 15;     // M row (A) / N col (B) within tile
    const int kh   = lane >> 4;     // 0 or 1 -> K offset 2*kh

    const int n0 = blockIdx.x * 64;
    const int m0 = (blockIdx.y * 8 + wave) * 16;
    const int b  = m0 >> 12;        // m0 / T   (T = 4096, tiles never cross b)
    const int t0 = m0 & (TDIM - 1);

    // A: x[b][k][t] at ((b*Z + k)*T + t); lane covers k = kk + 2*kh + {0,1}
    const float* aptr = x + ((size_t)b * ZDIM + 2 * kh) * TDIM + (size_t)(t0 + mr);
    // B: W[n][k]; lane n = n0 + j*16 + mr; contiguous k pair at 2*kh
    const float* bptr = w + (size_t)(n0 + mr) * ZDIM + 2 * kh;

    v8f acc0 = {}, acc1 = {}, acc2 = {}, acc3 = {};

#pragma unroll 8
    for (int kk = 0; kk < ZDIM; kk += 4) {
        v2f a;
        a.x = aptr[(size_t)kk * TDIM];
        a.y = aptr[(size_t)kk * TDIM + TDIM];
        v2f b0 = *(const v2f*)(bptr + kk);
        v2f b1 = *(const v2f*)(bptr + 16 * ZDIM + kk);
        v2f b2 = *(const v2f*)(bptr + 32 * ZDIM + kk);
        v2f b3 = *(const v2f*)(bptr + 48 * ZDIM + kk);
        acc0 = __builtin_amdgcn_wmma_f32_16x16x4_f32(false, a, false, b0, (short)0, acc0, false, false);
        acc1 = __builtin_amdgcn_wmma_f32_16x16x4_f32(false, a, false, b1, (short)0, acc1, false, false);
        acc2 = __builtin_amdgcn_wmma_f32_16x16x4_f32(false, a, false, b2, (short)0, acc2, false, false);
        acc3 = __builtin_amdgcn_wmma_f32_16x16x4_f32(false, a, false, b3, (short)0, acc3, false, false);
    }

    // Epilogue: C/D layout -> lane n = (lane&15), VGPR r holds M = r + 8*(lane>>4)
    const float bv0 = bias[n0 +      mr];
    const float bv1 = bias[n0 + 16 + mr];
    const float bv2 = bias[n0 + 32 + mr];
    const float bv3 = bias[n0 + 48 + mr];

#pragma unroll
    for (int r = 0; r < 8; ++r) {
        const size_t row = (size_t)(m0 + 8 * kh + r) * GE_DIM + n0;
        logits[row +      mr] = acc0[r] + bv0;
        logits[row + 16 + mr] = acc1[r] + bv1;
        logits[row + 32 + mr] = acc2[r] + bv2;
        logits[row + 48 + mr] = acc3[r] + bv3;
    }
}

// ---------------------------------------------------------------------------
// Kernel 2: per (b,t,g) row: idx = argmax(logits + gumbel) over 320 entries,
//           then q[bt, g*128:(g+1)*128] = codevectors[g, idx, :].
// One wave32 per row; xor-shuffle reduction (first-max tie-break, matching
// jnp.argmax); 32 lanes copy the 128-float codevector as float4 (b128).
// ---------------------------------------------------------------------------
__global__ __launch_bounds__(256) void pq_argmax_gather(
    const float* __restrict__ logits, const float* __restrict__ gumbel,
    const float* __restrict__ cv, float* __restrict__ q)
{
    const int lane = threadIdx.x & 31;
    const int wid  = blockIdx.x * (blockDim.x >> 5) + (threadIdx.x >> 5);
    const int g    = wid & 1;
    const int bt   = wid >> 1;

    const float* lp = logits + (size_t)bt * GE_DIM + (size_t)g * NENT;
    const float* gp = gumbel + (size_t)wid * NENT;

    float best = -3.402823466e38f;
    int   bi   = 0;
#pragma unroll
    for (int e = lane; e < NENT; e += 32) {
        const float v = lp[e] + gp[e];
        if (v > best) { best = v; bi = e; }
    }
#pragma unroll
    for (int off = 16; off > 0; off >>= 1) {
        const float ov = __shfl_xor(best, off, 32);
        const int   oi = __shfl_xor(bi,   off, 32);
        if (ov > best || (ov == best && oi < bi)) { best = ov; bi = oi; }
    }

    const float4* src = (const float4*)(cv + ((size_t)g * NENT + (size_t)bi) * VDIM);
    float4*       dst = (float4*)(q + (size_t)bt * (NGRP * VDIM) + (size_t)g * VDIM);
    dst[lane] = src[lane];
}

// ---------------------------------------------------------------------------
extern "C" void kernel_launch(void* const* d_in, const int* in_sizes, int n_in,
                              void* d_out, int out_size, void* d_ws, size_t ws_size,
                              hipStream_t stream)
{
    const float* x    = (const float*)d_in[0];  // (16, 512, 4096)
    const float* gn   = (const float*)d_in[1];  // (16, 4096, 2, 320)
    const float* w    = (const float*)d_in[2];  // (640, 512)
    const float* bias = (const float*)d_in[3];  // (640,)
    const float* cv   = (const float*)d_in[4];  // (1, 640, 128)

    float* logits = (float*)d_out;                         // 65536*640 floats
    float* q      = logits + (size_t)MTOT * GE_DIM;        // 65536*256 floats

    dim3 grid1(GE_DIM / 64, MTOT / (16 * 8));              // (10, 512)
    pq_gemm_logits<<<grid1, 256, 0, stream>>>(x, w, bias, logits);

    const int nwaves = MTOT * NGRP;                        // 131072 rows
    pq_argmax_gather<<<dim3(nwaves / 8), 256, 0, stream>>>(logits, gn, cv, q);
}